// SimpleBEVSampling_20675972563512
// MI455X (gfx1250) — compile-verified
//
#include <hip/hip_runtime.h>

// ---------------------------------------------------------------------------
// SimpleBEVSampling for MI455X (gfx1250, wave32)
//
// Phase 1 (per wave): project 16 (q,p) points through all 6 lidar2img
//   matrices using V_WMMA_F32_16X16X4_F32 (K=4 == homogeneous coords).
//   A (16x4)  : 16 points x (x,y,z,1)        -> 2 VGPRs/lane
//   B (4x16)  : column n = cam(n>>2), row(n&3); WMMA#1 = cams 0..3,
//               WMMA#2 cols 0..7 = cams 4..5 (cols 8..15 duplicate cams,
//               results land in unread LDS columns -> no masking needed)
//   D (16x16) : cam coords, spilled to LDS for uniform broadcast reads.
//
// Phase 2 (per wave): for each of its 16 points, loop cameras; mask is
//   wave-uniform -> whole camera skipped via scalar branch when invisible.
//   Channels mapped across lanes (c = lane + 32*j, j=0..3); bilinear taps
//   are 4-byte gathers that hit L2 (feats total only ~10.8 MB << 192 MB L2).
//   Output written with non-temporal stores (pure streaming, 33.5 MB).
// ---------------------------------------------------------------------------

typedef __attribute__((ext_vector_type(2))) float v2f;
typedef __attribute__((ext_vector_type(8))) float v8f;

#define NCAM 6
#define CCH 128
#define PTS_PER_WAVE 16
#define WAVES_PER_BLOCK 4
#define LDS_ROW 33     // 32 used (cams 0..5 in cols 0..23, scratch 24..31), +1 pad
#define M_TOTAL 65536  // Q*P = 16384*4

__device__ __forceinline__ void sample_level(const float* __restrict__ base,
                                             int Wl, int Hl, float u, float v,
                                             int lane, float acc[4]) {
  // grid_sample(bilinear, zeros, align_corners=False):
  // pixel coord = u*W - 0.5 (u in (0,1) guaranteed by mask)
  float x = u * (float)Wl - 0.5f;
  float y = v * (float)Hl - 0.5f;
  float x0f = floorf(x), y0f = floorf(y);
  int x0 = (int)x0f, y0 = (int)y0f;
  int x1 = x0 + 1, y1 = y0 + 1;
  float wx1 = x - x0f, wx0 = 1.0f - wx1;
  float wy1 = y - y0f, wy0 = 1.0f - wy1;
  // x0 in [-1, W-1], x1 in [0, W]; only these edges can be invalid.
  float vx0 = (x0 >= 0) ? 1.0f : 0.0f;
  float vx1 = (x1 <= Wl - 1) ? 1.0f : 0.0f;
  float vy0 = (y0 >= 0) ? 1.0f : 0.0f;
  float vy1 = (y1 <= Hl - 1) ? 1.0f : 0.0f;
  float w00 = wx0 * wy0 * vx0 * vy0;
  float w01 = wx1 * wy0 * vx1 * vy0;
  float w10 = wx0 * wy1 * vx0 * vy1;
  float w11 = wx1 * wy1 * vx1 * vy1;
  int xc0 = (x0 < 0) ? 0 : x0;
  int xc1 = (x1 > Wl - 1) ? (Wl - 1) : x1;
  int yc0 = (y0 < 0) ? 0 : y0;
  int yc1 = (y1 > Hl - 1) ? (Hl - 1) : y1;
  int o00 = yc0 * Wl + xc0;
  int o01 = yc0 * Wl + xc1;
  int o10 = yc1 * Wl + xc0;
  int o11 = yc1 * Wl + xc1;
  int hw = Hl * Wl;
#pragma unroll
  for (int j = 0; j < 4; ++j) {
    const float* pc = base + (size_t)(lane + j * 32) * (size_t)hw;
    float s = w00 * pc[o00] + w01 * pc[o01] + w10 * pc[o10] + w11 * pc[o11];
    acc[j] += 0.5f * s;  // (level0 + level1) / 2 folded into per-level 0.5
  }
}

__global__ __launch_bounds__(WAVES_PER_BLOCK * 32)
void bev_sample_kernel(const float* __restrict__ refpts,   // [4,128,128,3]
                       const float* __restrict__ feats0,   // [6,128,32,88]
                       const float* __restrict__ feats1,   // [6,128,16,44]
                       const float* __restrict__ l2i,      // [6,4,4]
                       float* __restrict__ out) {          // [16384,4,128]
  __shared__ float lds_cam[WAVES_PER_BLOCK][PTS_PER_WAVE][LDS_ROW];

  const int tid  = threadIdx.x;
  const int lane = tid & 31;
  const int w    = tid >> 5;
  const int half = lane >> 4;   // lane half selects K-pair {0,1} vs {2,3}
  const int mi   = lane & 15;   // A: row M ; B/D: column N
  const int m0   = (blockIdx.x * WAVES_PER_BLOCK + w) * PTS_PER_WAVE;

  // ------------------ Phase 1: WMMA projection ------------------
  {
    int m  = m0 + mi;          // flat point id = q*4 + p
    int p  = m & 3;
    int q  = m >> 2;
    int hb = q >> 7;
    int wb = q & 127;
    const float* rp = refpts + (((size_t)p * 128 + hb) * 128 + wb) * 3;
    // Branch-free component loads: lanes 0-15 need (x,y); 16-31 need (z,1).
    float v0 = rp[2 * half];     // x (half=0) or z (half=1)
    float v1 = rp[1];            // y (unused value for half=1)
    float scale = half ? 8.0f   : 102.4f;
    float bias  = half ? -5.0f  : -51.2f;

    // A 16x4 layout: VGPR0 = K=(half?2:0), VGPR1 = K+1 ; A = (x,y,z,1)
    v2f a;
    a.x = v0 * scale + bias;
    a.y = half ? 1.0f : (v1 * 102.4f - 51.2f);

    // B 4x16 layout: lane holds column mi, rows K=(half*2, half*2+1).
    // WMMA#2: cols 0..7 = cams 4..5; cols 8..15 duplicate them (harmless,
    // their outputs go to LDS columns 24..31 which are never read).
    int k0 = half * 2;
    v2f b1, b2;
    b1.x = l2i[mi * 4 + k0];            // flat (cam*4+row)*4 + k
    b1.y = l2i[mi * 4 + k0 + 1];
    int mi2 = mi & 7;
    b2.x = l2i[64 + mi2 * 4 + k0];
    b2.y = l2i[64 + mi2 * 4 + k0 + 1];

    v8f cz = {};
    v8f d1 = __builtin_amdgcn_wmma_f32_16x16x4_f32(
        false, a, false, b1, (short)0, cz, false, false);
    v8f d2 = __builtin_amdgcn_wmma_f32_16x16x4_f32(
        false, a, false, b2, (short)0, cz, false, false);

    // D 16x16 layout: lane half-> M = r + half*8, N = mi.  All stores
    // unconditional -> straight-line ds_store, no exec juggling.
#pragma unroll
    for (int r = 0; r < 8; ++r) {
      int row = r + half * 8;
      lds_cam[w][row][mi]      = d1[r];   // cams 0..3 -> cols 0..15
      lds_cam[w][row][16 + mi] = d2[r];   // cams 4..5 -> cols 16..23 (+scratch)
    }
  }
  __syncthreads();

  // ------------------ Phase 2: masked bilinear gather ------------------
  for (int i = 0; i < PTS_PER_WAVE; ++i) {
    float acc[4] = {0.0f, 0.0f, 0.0f, 0.0f};
    float den = 0.0f;
#pragma unroll
    for (int n = 0; n < NCAM; ++n) {
      // broadcast reads (all lanes same address) -> conflict-free
      float cx  = lds_cam[w][i][n * 4 + 0];
      float cy  = lds_cam[w][i][n * 4 + 1];
      float czv = lds_cam[w][i][n * 4 + 2];
      float dz = fmaxf(czv, 1e-5f);
      float u = cx / (dz * 704.0f);
      float v = cy / (dz * 256.0f);
      bool mask = (czv > 1e-5f) & (u > 0.0f) & (u < 1.0f) &
                  (v > 0.0f) & (v < 1.0f);
      if (mask) {  // wave-uniform in practice -> scalar skip of camera
        den += 1.0f;
        sample_level(feats0 + (size_t)n * CCH * 32 * 88, 88, 32, u, v, lane, acc);
        sample_level(feats1 + (size_t)n * CCH * 16 * 44, 44, 16, u, v, lane, acc);
      }
    }
    float dd = fmaxf(den, 1.0f);
    size_t obase = (size_t)(m0 + i) * CCH + lane;
#pragma unroll
    for (int j = 0; j < 4; ++j) {
      __builtin_nontemporal_store(acc[j] / dd, &out[obase + (size_t)j * 32]);
    }
  }
}

extern "C" void kernel_launch(void* const* d_in, const int* in_sizes, int n_in,
                              void* d_out, int out_size, void* d_ws, size_t ws_size,
                              hipStream_t stream) {
  (void)in_sizes; (void)n_in; (void)out_size; (void)d_ws; (void)ws_size;
  const float* refpts = (const float*)d_in[0];   // (1,4,128,128,3) f32
  const float* feats0 = (const float*)d_in[1];   // (1,6,128,32,88) f32
  const float* feats1 = (const float*)d_in[2];   // (1,6,128,16,44) f32
  const float* l2i    = (const float*)d_in[3];   // (1,6,4,4) f32
  float* out = (float*)d_out;                    // (1,16384,4,128) f32

  const int blocks = M_TOTAL / (WAVES_PER_BLOCK * PTS_PER_WAVE);  // 1024
  bev_sample_kernel<<<blocks, WAVES_PER_BLOCK * 32, 0, stream>>>(
      refpts, feats0, feats1, l2i, out);
}